// DySample_14748917694704
// MI455X (gfx1250) — compile-verified
//
#include <hip/hip_runtime.h>

typedef __attribute__((ext_vector_type(2))) float v2f;
typedef __attribute__((ext_vector_type(8))) float v8f;

#define Bsz   16
#define Cch   64
#define Hh    128
#define Ww    128
#define HW    (Hh * Ww)      // 16384 input plane
#define OUTW  (2 * Ww)       // 256
#define OUTHW (OUTW * OUTW)  // 65536 output plane

// Fused DySample: 1x1 conv (64->32) via V_WMMA_F32_16X16X4_F32, then in-register
// bilinear grid sampling with coalesced 8-byte non-temporal stores.
// One wave handles one 16-pixel tile (pixels share a row).
__global__ __launch_bounds__(256) void dysample_fused(
    const float* __restrict__ x, const float* __restrict__ w_off,
    const float* __restrict__ b_off, float* __restrict__ out)
{
  const int lane = threadIdx.x & 31;
  const int wv   = (blockIdx.x * blockDim.x + threadIdx.x) >> 5;  // 0..16383
  const int n    = lane & 15;   // pixel within tile / N column / M row
  const int kh   = lane >> 4;   // lane-half selects K pair & D-row half

  // ---- A fragments (weights), identical for every tile ------------------
  // f32 16x4 A layout: lane = mrow + 16*(kk>>1), vgpr = kk&1
  // -> lane(n,kh): vgpr v holds w_off[Mtile*16 + n][4k + 2*kh + v]
  v2f A0[16], A1[16];
  {
    const float* wp0 = w_off + n * 64        + kh * 2;
    const float* wp1 = w_off + (16 + n) * 64 + kh * 2;
#pragma unroll
    for (int k = 0; k < 16; ++k) {
      A0[k].x = wp0[k * 4 + 0]; A0[k].y = wp0[k * 4 + 1];
      A1[k].x = wp1[k * 4 + 0]; A1[k].y = wp1[k * 4 + 1];
    }
  }

  // ---- tile coordinates --------------------------------------------------
  const int p0  = wv << 4;           // flat pixel over B*H*W
  const int b   = p0 >> 14;          // / 16384
  const int hw0 = p0 & (HW - 1);
  const int h   = hw0 >> 7;
  const int w   = (hw0 & (Ww - 1)) + n;

  const float* xb   = x + (size_t)b * Cch * HW;  // batch base
  const float* xpix = xb + hw0 + n;              // + ch*HW per channel

  // ---- GEMM: acc0 = offset ch 0..15 (x), acc1 = ch 16..31 (y) ------------
  // f32 4x16 B layout: lane = n + 16*(kk>>1), vgpr = kk&1
  v8f acc0 = {};
  v8f acc1 = {};
#pragma unroll
  for (int k = 0; k < 16; ++k) {
    const int ch0 = k * 4 + kh * 2;
    v2f bb;
    bb.x = xpix[(size_t)ch0 * HW];
    bb.y = xpix[(size_t)(ch0 + 1) * HW];
    acc0 = __builtin_amdgcn_wmma_f32_16x16x4_f32(false, A0[k], false, bb,
                                                 (short)0, acc0, false, false);
    acc1 = __builtin_amdgcn_wmma_f32_16x16x4_f32(false, A1[k], false, bb,
                                                 (short)0, acc1, false, false);
  }

  // ---- fused bilinear sampling ------------------------------------------
  // D layout: lane(n,kh) VGPR r <-> offset channel o = r + 8*kh (0..15)
  // o = g*4 + i*2 + j  (group g, sub-pixel i,j); y-offset channel = o+16.
  // grid_sample algebra collapses to ix = clamp(w + off_x), iy = clamp(h + off_y).
  float* outb = out + (size_t)b * Cch * OUTHW;

#pragma unroll
  for (int rp = 0; rp < 8; rp += 2) {
    const int o0 = rp + 8 * kh;        // j = 0
    const int o1 = o0 + 1;             // j = 1 (same g, same i)
    const int g  = o0 >> 2;
    const int a  = g & 1;
    const int i  = (o0 & 3) >> 1;

    // init_pos(o) == init_pos(o+16): sel = (a==0) ? j : i ; ip = sel*0.5-0.25
    const float ip0 = ((a == 0) ? 0.0f : (float)i) * 0.5f - 0.25f;
    const float ip1 = ((a == 0) ? 1.0f : (float)i) * 0.5f - 0.25f;

    const float offx0 = (acc0[rp]     + b_off[o0])      * 0.25f + ip0;
    const float offy0 = (acc1[rp]     + b_off[o0 + 16]) * 0.25f + ip0;
    const float offx1 = (acc0[rp + 1] + b_off[o1])      * 0.25f + ip1;
    const float offy1 = (acc1[rp + 1] + b_off[o1 + 16]) * 0.25f + ip1;

    // sample 0 (j=0)
    float ix0 = fminf(fmaxf((float)w + offx0, 0.0f), 127.0f);
    float iy0 = fminf(fmaxf((float)h + offy0, 0.0f), 127.0f);
    float xf0 = floorf(ix0), yf0 = floorf(iy0);
    float fx0 = ix0 - xf0,  fy0 = iy0 - yf0;
    int   x00 = (int)xf0,   y00 = (int)yf0;
    int   x01 = min(x00 + 1, Ww - 1), y01 = min(y00 + 1, Hh - 1);
    float s0w00 = (1.0f - fy0) * (1.0f - fx0);
    float s0w01 = (1.0f - fy0) * fx0;
    float s0w10 = fy0 * (1.0f - fx0);
    float s0w11 = fy0 * fx0;
    int s0i00 = y00 * Ww + x00, s0i01 = y00 * Ww + x01;
    int s0i10 = y01 * Ww + x00, s0i11 = y01 * Ww + x01;

    // sample 1 (j=1)
    float ix1 = fminf(fmaxf((float)w + offx1, 0.0f), 127.0f);
    float iy1 = fminf(fmaxf((float)h + offy1, 0.0f), 127.0f);
    float xf1 = floorf(ix1), yf1 = floorf(iy1);
    float fx1 = ix1 - xf1,  fy1 = iy1 - yf1;
    int   x10 = (int)xf1,   y10 = (int)yf1;
    int   x11 = min(x10 + 1, Ww - 1), y11 = min(y10 + 1, Hh - 1);
    float s1w00 = (1.0f - fy1) * (1.0f - fx1);
    float s1w01 = (1.0f - fy1) * fx1;
    float s1w10 = fy1 * (1.0f - fx1);
    float s1w11 = fy1 * fx1;
    int s1i00 = y10 * Ww + x10, s1i01 = y10 * Ww + x11;
    int s1i10 = y11 * Ww + x10, s1i11 = y11 * Ww + x11;

    const float* srcg = xb + (size_t)(g * 16) * HW;
    float* dstbase = outb + (size_t)(g * 16) * OUTHW
                   + (size_t)(2 * h + i) * OUTW + 2 * w;  // j packed in v2f

#pragma unroll
    for (int cc = 0; cc < 16; ++cc) {
      const float* p = srcg + (size_t)cc * HW;
      float v0 = s0w00 * p[s0i00] + s0w01 * p[s0i01]
               + s0w10 * p[s0i10] + s0w11 * p[s0i11];
      float v1 = s1w00 * p[s1i00] + s1w01 * p[s1i01]
               + s1w10 * p[s1i10] + s1w11 * p[s1i11];
      v2f vv;
      vv.x = v0;
      vv.y = v1;
      // output is streamed once (268 MB) -> non-temporal, keep x hot in L2
      __builtin_nontemporal_store(vv, (v2f*)(dstbase + (size_t)cc * OUTHW));
    }
  }
}

extern "C" void kernel_launch(void* const* d_in, const int* in_sizes, int n_in,
                              void* d_out, int out_size, void* d_ws, size_t ws_size,
                              hipStream_t stream) {
  const float* x     = (const float*)d_in[0];  // [16,64,128,128]
  const float* w_off = (const float*)d_in[1];  // [32,64]
  const float* b_off = (const float*)d_in[2];  // [32]
  float* out = (float*)d_out;                  // [16,64,256,256]

  // 16384 pixel tiles of 16 -> one per wave; 8 waves/block -> 2048 blocks
  dim3 grid(2048), block(256);
  hipLaunchKernelGGL(dysample_fused, grid, block, 0, stream,
                     x, w_off, b_off, out);
}